// ALiBi_23785528885810
// MI455X (gfx1250) — compile-verified
//
#include <hip/hip_runtime.h>
#include <stdint.h>

// Problem constants (reference: B=2, H=16, S=2048, CTX=4096)
#define NB   2
#define NH   16
#define NS   2048
#define ROWS 8           // rows of the [S,S] matrix handled per block
#define TPB  256         // threads per block (8 wave32s)

typedef float v4f __attribute__((ext_vector_type(4)));
typedef int   v4i __attribute__((ext_vector_type(4)));

// Exact pointer types the async builtin wants (per hipcc diagnostic):
// param0: v4i in address_space(1) ("__device__"), param1: v4i in AS(3).
typedef __attribute__((address_space(1))) v4i g_v4i;
typedef __attribute__((address_space(3))) v4i l_v4i;

// ---------------------------------------------------------------------------
// gfx1250 async copy: global -> LDS, 128-bit per lane, tracked by ASYNCcnt.
// ---------------------------------------------------------------------------
__device__ __forceinline__ void async_b128(const float* gsrc, uint32_t lds_off) {
#if __has_builtin(__builtin_amdgcn_global_load_async_to_lds_b128)
    __builtin_amdgcn_global_load_async_to_lds_b128(
        (g_v4i*)(uintptr_t)gsrc,
        (l_v4i*)(uintptr_t)lds_off,
        /*offset=*/0, /*cpol=*/0);
#else
    asm volatile("global_load_async_to_lds_b128 %0, %1, off"
                 :: "v"(lds_off), "v"(gsrc)
                 : "memory");
#endif
}

__device__ __forceinline__ void wait_async0() {
#if __has_builtin(__builtin_amdgcn_s_wait_asynccnt)
    __builtin_amdgcn_s_wait_asynccnt(0);
#else
    asm volatile("s_wait_asynccnt 0" ::: "memory");
#endif
}

// out = sc - slope*(pi - pj), componentwise, with the reference's
// mul-then-sub rounding (no FMA contraction).
__device__ __forceinline__ v4f apply4(v4f sc, v4f pj, float pi, float slope) {
    v4f r;
    r.x = __fsub_rn(sc.x, __fmul_rn(slope, __fsub_rn(pi, pj.x)));
    r.y = __fsub_rn(sc.y, __fmul_rn(slope, __fsub_rn(pi, pj.y)));
    r.z = __fsub_rn(sc.z, __fmul_rn(slope, __fsub_rn(pi, pj.z)));
    r.w = __fsub_rn(sc.w, __fmul_rn(slope, __fsub_rn(pi, pj.w)));
    return r;
}

// ---------------------------------------------------------------------------
// Kernel 1: pos_ws[b*S+s] = positions[token_indices[b*S+s]]  (4096 gathers)
// ---------------------------------------------------------------------------
__global__ __launch_bounds__(TPB) void alibi_gather_pos(
    const float* __restrict__ positions,
    const long long* __restrict__ tok,
    float* __restrict__ pos_ws) {
    int idx = blockIdx.x * TPB + threadIdx.x;
    if (idx < NB * NS) {
        pos_ws[idx] = positions[(int)tok[idx]];
    }
}

// ---------------------------------------------------------------------------
// Kernel 2: streaming bias-subtract. One block handles ROWS consecutive rows
// of one (b,h) slice. pos[b,:] (8 KB) is async-staged into LDS once per
// block; pos_j columns are hoisted to registers and reused across rows.
// ---------------------------------------------------------------------------
__global__ __launch_bounds__(TPB) void alibi_apply(
    const float* __restrict__ scores,
    const float* __restrict__ pos_ws,
    float* __restrict__ out) {
    __shared__ v4f spos4[NS / 4];   // 8 KB: pos[b, 0..S)

    const int tid  = threadIdx.x;
    const int row0 = blockIdx.x * ROWS;       // flat row id in [0, B*H*S)
    const int b    = row0 / (NH * NS);
    const int h    = (row0 / NS) % NH;
    const int i0   = row0 % NS;               // ROWS consecutive i share b,h

    // slope = 2^(-(h+1)/2), built exactly (matches correctly-rounded exp2)
    const int   hp    = h + 1;
    const float slope = ldexpf((hp & 1) ? 0.70710678118654752440f : 1.0f,
                               -(hp >> 1));

    // Stage pos[b,:] into LDS with CDNA5 async loads (ASYNCcnt-tracked).
    const float* gpos = pos_ws + (size_t)b * NS;
    for (int k = tid; k < NS / 4; k += TPB) {
        async_b128(gpos + 4 * k, (uint32_t)(uintptr_t)&spos4[k]);
    }
    wait_async0();        // each wave drains its own async copies...
    __syncthreads();      // ...then the barrier makes the full tile visible

    const float* sposf = (const float*)spos4;
    // Each thread owns two fixed float4 column groups, reused for all rows.
    const v4f pj0 = spos4[tid];
    const v4f pj1 = spos4[tid + TPB];

    const v4f* s4 = (const v4f*)(scores + (size_t)row0 * NS);
    v4f*       o4 = (v4f*)(out + (size_t)row0 * NS);

#pragma unroll
    for (int r = 0; r < ROWS; ++r) {
        const float  pi   = sposf[i0 + r];              // LDS broadcast read
        const size_t base = (size_t)r * (NS / 4);

        v4f sc0 = __builtin_nontemporal_load(s4 + base + tid);
        v4f sc1 = __builtin_nontemporal_load(s4 + base + tid + TPB);

        v4f r0 = apply4(sc0, pj0, pi, slope);
        v4f r1 = apply4(sc1, pj1, pi, slope);

        __builtin_nontemporal_store(r0, o4 + base + tid);
        __builtin_nontemporal_store(r1, o4 + base + tid + TPB);
    }
}

// ---------------------------------------------------------------------------
extern "C" void kernel_launch(void* const* d_in, const int* in_sizes, int n_in,
                              void* d_out, int out_size, void* d_ws, size_t ws_size,
                              hipStream_t stream) {
    const float*     scores    = (const float*)d_in[0];
    const float*     positions = (const float*)d_in[1];
    const long long* tok       = (const long long*)d_in[2];
    float*           out       = (float*)d_out;
    float*           pos_ws    = (float*)d_ws;   // needs NB*NS*4 = 16 KB

    alibi_gather_pos<<<(NB * NS + TPB - 1) / TPB, TPB, 0, stream>>>(
        positions, tok, pos_ws);

    alibi_apply<<<(NB * NH * NS) / ROWS, TPB, 0, stream>>>(
        scores, pos_ws, out);
}